// UserEncoder_71811853189246
// MI455X (gfx1250) — compile-verified
//
#include <hip/hip_runtime.h>

typedef __attribute__((ext_vector_type(2))) float v2f;
typedef __attribute__((ext_vector_type(8))) float v8f;

#define BN   1024
#define CC   64
#define HH   64
#define DD   256
#define HIDN 256
#define HSTR 264   // hist LDS row stride (floats): 2*HSTR % 64 == 16 -> half-waves hit disjoint banks
#define WSTR 65    // weights LDS row stride (floats): row-major column reads spread across banks

// ---------------------------------------------------------------------------
// Kernel 0: collapse the two Linears:  v_c = W1[:, :D]^T w2,  v_h = W1[:, D:]^T w2,
//           bias = w2 . b1 + b2.   Tiny (0.26 MFLOP), one block.
// ---------------------------------------------------------------------------
__global__ void __launch_bounds__(256)
precompute_vec_kernel(const float* __restrict__ W1,   // [HID, 2D]
                      const float* __restrict__ b1,   // [HID]
                      const float* __restrict__ W2,   // [1, HID]
                      const float* __restrict__ b2,   // [1]
                      float* __restrict__ vout)       // vc[256] vh[256] bias
{
    const int d = threadIdx.x;            // 0..255
    float ac = 0.f, ah = 0.f;
    for (int k = 0; k < HIDN; ++k) {
        const float w2k = W2[k];
        ac += W1[k * (2 * DD) + d]      * w2k;
        ah += W1[k * (2 * DD) + DD + d] * w2k;
    }
    vout[d]      = ac;
    vout[DD + d] = ah;
    if (d == 0) {
        float bb = b2[0];
        for (int k = 0; k < HIDN; ++k) bb += W2[k] * b1[k];
        vout[2 * DD] = bb;
    }
}

// ---------------------------------------------------------------------------
// Kernel 1: one workgroup per batch element b (1024 blocks x 256 threads).
//   phase 1: stage hist[b] (64x256 f32) -> LDS (padded stride)
//   phase 2: s_h (LDS dot v_h), s_c (global dot v_c)
//   phase 3: masked softmax rows -> weights LDS tile [64x65]
//   phase 4: out[b] = weights(64x64) x hist(64x256) via v_wmma_f32_16x16x4_f32
// ---------------------------------------------------------------------------
__global__ void __launch_bounds__(256)
user_encoder_kernel(const float* __restrict__ cand,            // [B,C,D]
                    const float* __restrict__ hist,            // [B,H,D]
                    const unsigned char* __restrict__ mask_c,  // [B,C] bool
                    const unsigned char* __restrict__ mask_h,  // [B,H] bool
                    const float* __restrict__ vcvh,            // vc,vh,bias
                    float* __restrict__ out)                   // [B,C,D]
{
    extern __shared__ float smem[];
    float* hs  = smem;                    // HH * HSTR   (hist tile)
    float* wl  = hs + HH * HSTR;          // CC * WSTR   (softmax weights)
    float* shl = wl + CC * WSTR;          // HH
    float* scl = shl + HH;                // CC

    const int tid = threadIdx.x;
    const int b   = blockIdx.x;
    const float* histB = hist + (size_t)b * HH * DD;
    const float* candB = cand + (size_t)b * CC * DD;

    // ---- phase 1: global -> LDS, float4 granularity, coalesced -------------
    for (int i = tid; i < HH * (DD / 4); i += 256) {
        const int h  = i >> 6;            // 64 float4 per row
        const int d4 = i & 63;
        const float4 v = ((const float4*)(histB + h * DD))[d4];
        *((float4*)(hs + h * HSTR + d4 * 4)) = v;   // (h*HSTR+d4*4)*4 bytes is 16B aligned
    }
    __syncthreads();

    const float biasv = vcvh[2 * DD];

    // ---- phase 2: scoring dots --------------------------------------------
    if (tid < HH) {                       // s_h[h] = hist[b,h,:] . v_h
        const int h = tid;
        const float* row = hs + h * HSTR;
        float acc = 0.f;
        for (int d = 0; d < DD; d += 4) {
            acc += row[d]     * vcvh[DD + d];
            acc += row[d + 1] * vcvh[DD + d + 1];
            acc += row[d + 2] * vcvh[DD + d + 2];
            acc += row[d + 3] * vcvh[DD + d + 3];
        }
        shl[h] = acc;
    } else if (tid < HH + CC) {           // s_c[c] = cand[b,c,:] . v_c
        const int c = tid - HH;
        const float* row = candB + c * DD;
        float acc = 0.f;
        for (int d = 0; d < DD; d += 4) {
            const float4 r = ((const float4*)row)[d >> 2];
            acc += r.x * vcvh[d] + r.y * vcvh[d + 1]
                 + r.z * vcvh[d + 2] + r.w * vcvh[d + 3];
        }
        scl[c] = acc;
    }
    __syncthreads();

    // ---- phase 3: masked softmax over h, one thread per candidate row ------
    if (tid < CC) {
        const int c = tid;
        const bool cm = mask_c[(size_t)b * CC + c] != 0;
        const float sc = scl[c] + biasv;
        float mx = -3.402823466e38f;
        for (int h = 0; h < HH; ++h)
            if (mask_h[(size_t)b * HH + h]) mx = fmaxf(mx, sc + shl[h]);
        float sum = 0.f;
        for (int h = 0; h < HH; ++h) {
            float e = 0.f;
            if (mask_h[(size_t)b * HH + h]) e = __expf(sc + shl[h] - mx);
            wl[c * WSTR + h] = e;
            sum += e;
        }
        const float inv = cm ? (1.f / sum) : 0.f;
        for (int h = 0; h < HH; ++h) wl[c * WSTR + h] *= inv;
    }
    __syncthreads();

    // ---- phase 4: out[b] = W(64x64) x Hist(64x256) via fp32 WMMA -----------
    // 4 M-tiles x 16 N-tiles of 16x16; wave w -> mTile = w&3, nGroup = w>>2.
    const int lane  = tid & 31;
    const int wave  = tid >> 5;
    const int mTile = wave & 3;
    const int nGrp  = wave >> 2;              // 0 or 1 (8 N-tiles each)
    const int mrow  = mTile * 16 + (lane & 15);
    const int koff  = (lane >> 4) * 2;        // A/B lane-half K offset: 0 or 2

    // A fragments (weights) are invariant across the N loop: hoist to VGPRs.
    v2f afrag[16];
#pragma unroll
    for (int k = 0; k < 16; ++k) {
        afrag[k].x = wl[mrow * WSTR + k * 4 + koff];
        afrag[k].y = wl[mrow * WSTR + k * 4 + koff + 1];
    }

    float* outB = out + (size_t)b * CC * DD;
#pragma unroll
    for (int j = 0; j < 8; ++j) {
        const int n = (nGrp * 8 + j) * 16 + (lane & 15);
        v8f acc = {};
#pragma unroll
        for (int k = 0; k < 16; ++k) {
            v2f bfrag;
            bfrag.x = hs[(k * 4 + koff)     * HSTR + n];
            bfrag.y = hs[(k * 4 + koff + 1) * HSTR + n];
            acc = __builtin_amdgcn_wmma_f32_16x16x4_f32(
                false, afrag[k], false, bfrag, (short)0, acc, false, false);
        }
        // D layout: VGPR i -> rows M=i (lanes 0-15) and M=i+8 (lanes 16-31)
        const int cbase = mTile * 16 + ((lane >> 4) ? 8 : 0);
#pragma unroll
        for (int i = 0; i < 8; ++i)
            outB[(size_t)(cbase + i) * DD + n] = acc[i];
    }
}

// ---------------------------------------------------------------------------
extern "C" void kernel_launch(void* const* d_in, const int* in_sizes, int n_in,
                              void* d_out, int out_size, void* d_ws, size_t ws_size,
                              hipStream_t stream) {
    const float*         cand = (const float*)d_in[0];
    const float*         hist = (const float*)d_in[1];
    const unsigned char* mc   = (const unsigned char*)d_in[2];
    const unsigned char* mh   = (const unsigned char*)d_in[3];
    const float*         W1   = (const float*)d_in[4];
    const float*         b1   = (const float*)d_in[5];
    const float*         W2   = (const float*)d_in[6];
    const float*         b2   = (const float*)d_in[7];
    float* vcvh = (float*)d_ws;           // 513 floats of scratch
    float* outp = (float*)d_out;

    precompute_vec_kernel<<<1, 256, 0, stream>>>(W1, b1, W2, b2, vcvh);

    const size_t smem = (size_t)(HH * HSTR + CC * WSTR + HH + CC) * sizeof(float);
    user_encoder_kernel<<<BN, 256, smem, stream>>>(cand, hist, mc, mh, vcvh, outp);
}